// EdgeAwareGraphSelfAttention_42348377538826
// MI455X (gfx1250) — compile-verified
//
#include <hip/hip_runtime.h>
#include <float.h>
#include <stddef.h>

// ---------------------------------------------------------------------------
// EdgeAwareGraphSelfAttention, fused for MI455X (gfx1250, wave32, WMMA f32).
//
// B=2, N=512, QD=512, ED=11, H=4, DH=64, INNER=256
// Key fusion: sum_j attn_ij * (gelu(e_ij@Wev1+bev1)@Wev2 + bev2)
//           = (sum_j attn_ij * gelu(e_ij@Wev1+bev1)) @ Wev2 + bev2
// which removes the [B,N,N,256]@[256,256] GEMM (68.7 GFLOP / 537 MB temp).
// ---------------------------------------------------------------------------

#define BB    2
#define NN_   512
#define QD_   512
#define ED_   11
#define H_    4
#define DH_   64
#define INNER 256
#define SCALE 0.125f  // 64^-0.5

typedef float v2f __attribute__((ext_vector_type(2)));
typedef float v8f __attribute__((ext_vector_type(8)));

typedef __attribute__((address_space(1))) int as1_int;  // global
typedef __attribute__((address_space(3))) int as3_int;  // LDS

__device__ __forceinline__ float gelu_exact(float v) {
    return 0.5f * v * (1.0f + erff(v * 0.70710678118654752440f));
}

#if defined(__has_builtin)
#if __has_builtin(__builtin_amdgcn_global_load_async_to_lds_b32)
#define HAVE_ASYNC_LDS 1
#endif
#endif

__device__ __forceinline__ void wait_asynccnt0() {
#if defined(__has_builtin) && __has_builtin(__builtin_amdgcn_s_wait_asynccnt)
    __builtin_amdgcn_s_wait_asynccnt(0);
#else
    asm volatile("s_wait_asynccnt 0" ::: "memory");
#endif
}

// ---------------------------------------------------------------------------
// f32 WMMA GEMM: each wave computes a 16(M) x 64(N) strip; 4 waves per block
// cover a 64x64 tile (waves share B addresses -> WGP$ hits). A-fragment is
// reused across 4 WMMAs per k-step; 4 independent accumulator chains hide the
// WMMA->WMMA RAW hazard. M % 64 == 0, N % 64 == 0, K % 4 == 0.
// A-frag (16x4 f32): lanes 0-15 -> M=lane, K=k0+0/1; lanes 16-31 -> K=k0+2/3
// B-frag (4x16 f32): VGPR0 rows k0 / k0+2, VGPR1 rows k0+1 / k0+3
// C/D: VGPR r -> row m0+r (lanes 0-15) / m0+r+8 (lanes 16-31), col n+(lane&15)
// ---------------------------------------------------------------------------
__global__ void wmma_gemm_f32(const float* __restrict__ A, int lda,
                              const float* __restrict__ B, int ldb,
                              float* __restrict__ C, int ldc,
                              const float* __restrict__ bias,
                              const float* __restrict__ Cadd, int ldcadd,
                              int K) {
    const int wave = threadIdx.x >> 5;
    const int lane = threadIdx.x & 31;
    const int half = lane >> 4;       // 0: lanes 0-15, 1: lanes 16-31
    const int lr   = lane & 15;
    const int m0 = blockIdx.x * 64 + wave * 16;
    const int n0 = blockIdx.y * 64;

    v8f acc[4] = {{}, {}, {}, {}};
    const float* Arow = A + (size_t)(m0 + lr) * lda + 2 * half;

#pragma unroll 2
    for (int k0 = 0; k0 < K; k0 += 4) {
        v2f a;
        a.x = Arow[k0];
        a.y = Arow[k0 + 1];
        __builtin_prefetch(Arow + k0 + 32, 0, 0);  // global_prefetch_b8

        const float* B0 = B + (size_t)(k0 + 2 * half) * ldb + n0 + lr;
        const float* B1 = B0 + ldb;
#pragma unroll
        for (int t = 0; t < 4; ++t) {
            v2f b;
            b.x = B0[t * 16];
            b.y = B1[t * 16];
            acc[t] = __builtin_amdgcn_wmma_f32_16x16x4_f32(
                /*neg_a=*/false, a, /*neg_b=*/false, b,
                /*c_mod=*/(short)0, acc[t], /*reuse_a=*/false, /*reuse_b=*/false);
        }
    }

#pragma unroll
    for (int t = 0; t < 4; ++t) {
        const int col = n0 + t * 16 + lr;
        const float bv = bias ? bias[col] : 0.0f;
#pragma unroll
        for (int r = 0; r < 8; ++r) {
            const int row = m0 + r + 8 * half;
            float o = acc[t][r] + bv;
            if (Cadd) o += Cadd[(size_t)row * ldcadd + col];
            C[(size_t)row * ldc + col] = o;
        }
    }
}

// ---------------------------------------------------------------------------
// Per (b,i): edge-bias MLP + q.k logits + mask + softmax -> attn[b,h,i,:]
// 256 threads; thread handles edges j = tid and tid+256.
// ---------------------------------------------------------------------------
__global__ void edge_bias_softmax(const float* __restrict__ q,
                                  const float* __restrict__ k,
                                  const float* __restrict__ edge_attr,
                                  const unsigned char* __restrict__ mask,
                                  const float* __restrict__ Web1,
                                  const float* __restrict__ beb1,
                                  const float* __restrict__ Web2,
                                  const float* __restrict__ beb2,
                                  float* __restrict__ attn) {
    __shared__ float sW1[ED_ * INNER];   // Web1 [11,256]
    __shared__ float sb1[INNER];
    __shared__ float sW2[INNER * H_];    // Web2 [256,4]
    __shared__ float sb2[H_];
    __shared__ float sq[INNER];          // q row for (b,i)
    __shared__ float ssim[H_][NN_];      // masked logits
    __shared__ float red[256];

    const int tid = threadIdx.x;
    const int bi  = blockIdx.x;
    const int b   = bi / NN_;
    const int i   = bi - b * NN_;

    for (int idx = tid; idx < ED_ * INNER; idx += 256) sW1[idx] = Web1[idx];
    for (int idx = tid; idx < INNER * H_; idx += 256)  sW2[idx] = Web2[idx];
    sb1[tid] = beb1[tid];
    sq[tid]  = q[(size_t)bi * INNER + tid];
    if (tid < H_) sb2[tid] = beb2[tid];
    __syncthreads();

#pragma unroll
    for (int jj = 0; jj < 2; ++jj) {
        const int j = tid + jj * 256;
        // edge features for this edge
        float e[ED_];
        const float* ep = edge_attr + (((size_t)b * NN_ + i) * NN_ + j) * ED_;
#pragma unroll
        for (int r = 0; r < ED_; ++r) e[r] = ep[r];

        // bias MLP: gelu(e@Web1 + beb1) @ Web2 + beb2 (all LDS reads broadcast)
        float biash[H_];
#pragma unroll
        for (int h = 0; h < H_; ++h) biash[h] = sb2[h];
        for (int c = 0; c < INNER; ++c) {
            float a = sb1[c];
#pragma unroll
            for (int r = 0; r < ED_; ++r) a += e[r] * sW1[r * INNER + c];
            const float g = gelu_exact(a);
#pragma unroll
            for (int h = 0; h < H_; ++h) biash[h] += g * sW2[c * H_ + h];
        }

        // q_i . k_j per head (k row streamed, L2-resident; q broadcast via LDS)
        float dot[H_] = {0.f, 0.f, 0.f, 0.f};
        const float4* kr4 =
            reinterpret_cast<const float4*>(k + ((size_t)b * NN_ + j) * INNER);
#pragma unroll 8
        for (int c4 = 0; c4 < INNER / 4; ++c4) {
            const float4 kk = kr4[c4];
            const int h = c4 >> 4;  // 16 float4 per 64-wide head
            const float* qq = &sq[c4 * 4];
            dot[h] += qq[0] * kk.x + qq[1] * kk.y + qq[2] * kk.z + qq[3] * kk.w;
        }

        const bool ok = mask[((size_t)b * NN_ + i) * NN_ + j] != 0;
#pragma unroll
        for (int h = 0; h < H_; ++h)
            ssim[h][j] = ok ? (dot[h] * SCALE + biash[h]) : -FLT_MAX;
    }
    __syncthreads();

    // per-head softmax over 512 logits
    for (int h = 0; h < H_; ++h) {
        float m = fmaxf(ssim[h][tid], ssim[h][tid + 256]);
        red[tid] = m;
        __syncthreads();
        for (int s = 128; s > 0; s >>= 1) {
            if (tid < s) red[tid] = fmaxf(red[tid], red[tid + s]);
            __syncthreads();
        }
        const float mx = red[0];
        __syncthreads();

        const float p0 = __expf(ssim[h][tid] - mx);
        const float p1 = __expf(ssim[h][tid + 256] - mx);
        red[tid] = p0 + p1;
        __syncthreads();
        for (int s = 128; s > 0; s >>= 1) {
            if (tid < s) red[tid] += red[tid + s];
            __syncthreads();
        }
        const float inv = 1.0f / red[0];
        __syncthreads();  // protect red for next head

        float* arow = attn + (((size_t)b * H_ + h) * NN_ + i) * NN_;
        arow[tid]       = p0 * inv;
        arow[tid + 256] = p1 * inv;
    }
}

// ---------------------------------------------------------------------------
// Per (b,i): t[b,h,i,c] = sum_j attn[b,h,i,j] * gelu(e_ij @ Wev1 + bev1)[c]
// 256 threads; thread = channel c. Edge tile + attn tile staged in LDS via
// ASYNCcnt-tracked async loads when the toolchain exposes them.
// ---------------------------------------------------------------------------
__global__ void edge_value_acc(const float* __restrict__ edge_attr,
                               const float* __restrict__ attn,
                               const float* __restrict__ Wev1,
                               const float* __restrict__ bev1,
                               float* __restrict__ t) {
    __shared__ float se[64 * ED_];   // edge tile [64 j][11]
    __shared__ float sa[H_ * 64];    // attn tile [4 h][64 j]

    const int tid = threadIdx.x;     // channel c
    const int bi  = blockIdx.x;
    const int b   = bi / NN_;
    const int i   = bi - b * NN_;

    float w1[ED_];
#pragma unroll
    for (int r = 0; r < ED_; ++r) w1[r] = Wev1[r * INNER + tid];
    const float b1 = bev1[tid];

    const int h_own  = tid >> 6;
    const int jo_own = tid & 63;

    float tacc[H_] = {0.f, 0.f, 0.f, 0.f};

    for (int tile = 0; tile < NN_ / 64; ++tile) {
        const int j0 = tile * 64;
        __syncthreads();  // previous tile fully consumed

        const float* esrc =
            edge_attr + (((size_t)b * NN_ + i) * NN_ + j0) * ED_;  // 704 floats
        const float* asrc =
            attn + (((size_t)b * H_ + h_own) * NN_ + i) * NN_ + j0 + jo_own;

#ifdef HAVE_ASYNC_LDS
        for (int idx = tid; idx < 64 * ED_; idx += 256)
            __builtin_amdgcn_global_load_async_to_lds_b32(
                (as1_int*)(esrc + idx), (as3_int*)(se + idx), 0, 0);
        __builtin_amdgcn_global_load_async_to_lds_b32(
            (as1_int*)asrc, (as3_int*)(sa + tid), 0, 0);
        wait_asynccnt0();
#else
        for (int idx = tid; idx < 64 * ED_; idx += 256) se[idx] = esrc[idx];
        sa[tid] = *asrc;
#endif
        __syncthreads();

        for (int jo = 0; jo < 64; ++jo) {
            float a = b1;
#pragma unroll
            for (int r = 0; r < ED_; ++r) a += se[jo * ED_ + r] * w1[r];
            const float g = gelu_exact(a);
#pragma unroll
            for (int h = 0; h < H_; ++h) tacc[h] += g * sa[h * 64 + jo];
        }
    }

#pragma unroll
    for (int h = 0; h < H_; ++h)
        t[(((size_t)b * H_ + h) * NN_ + i) * INNER + tid] = tacc[h];
}

// ---------------------------------------------------------------------------
// Host launcher
// ---------------------------------------------------------------------------
extern "C" void kernel_launch(void* const* d_in, const int* in_sizes, int n_in,
                              void* d_out, int out_size, void* d_ws, size_t ws_size,
                              hipStream_t stream) {
    const float*         x    = (const float*)d_in[0];
    const unsigned char* msk  = (const unsigned char*)d_in[1];  // jnp bool -> u8
    const float*         ea   = (const float*)d_in[2];
    const float*         Wq   = (const float*)d_in[3];
    const float*         Wk   = (const float*)d_in[4];
    const float*         Wv   = (const float*)d_in[5];
    const float*         Web1 = (const float*)d_in[6];
    const float*         beb1 = (const float*)d_in[7];
    const float*         Web2 = (const float*)d_in[8];
    const float*         beb2 = (const float*)d_in[9];
    const float*         Wev1 = (const float*)d_in[10];
    const float*         bev1 = (const float*)d_in[11];
    const float*         Wev2 = (const float*)d_in[12];
    const float*         bev2 = (const float*)d_in[13];
    const float*         Wo   = (const float*)d_in[14];
    const float*         bo   = (const float*)d_in[15];

    float* out  = (float*)d_out;                          // [B,N,QD]
    float* attn = out + (size_t)BB * NN_ * QD_;           // [B,H,N,N]

    // workspace: q,k,v,hout (1 MB each) + t (4 MB) = 8.39 MB
    float* qb   = (float*)d_ws;
    float* kb   = qb + (size_t)BB * NN_ * INNER;
    float* vb   = kb + (size_t)BB * NN_ * INNER;
    float* hout = vb + (size_t)BB * NN_ * INNER;
    float* tb   = hout + (size_t)BB * NN_ * INNER;
    (void)in_sizes; (void)n_in; (void)out_size; (void)ws_size;

    const int M = BB * NN_;  // 1024

    // 1) QKV projections: [1024,512]@[512,256]  (WMMA)
    wmma_gemm_f32<<<dim3(M / 64, INNER / 64), 128, 0, stream>>>(
        x, QD_, Wq, INNER, qb, INNER, nullptr, nullptr, 0, QD_);
    wmma_gemm_f32<<<dim3(M / 64, INNER / 64), 128, 0, stream>>>(
        x, QD_, Wk, INNER, kb, INNER, nullptr, nullptr, 0, QD_);
    wmma_gemm_f32<<<dim3(M / 64, INNER / 64), 128, 0, stream>>>(
        x, QD_, Wv, INNER, vb, INNER, nullptr, nullptr, 0, QD_);

    // 2) logits (q.k + edge-bias MLP) + mask + softmax -> attn
    edge_bias_softmax<<<BB * NN_, 256, 0, stream>>>(
        qb, kb, ea, msk, Web1, beb1, Web2, beb2, attn);

    // 3a) attn @ v per (b,h): [512,512]@[512,64]  (WMMA)
    for (int b = 0; b < BB; ++b)
        for (int h = 0; h < H_; ++h)
            wmma_gemm_f32<<<dim3(NN_ / 64, DH_ / 64), 128, 0, stream>>>(
                attn + (((size_t)b * H_ + h) * NN_) * NN_, NN_,
                vb + (size_t)b * NN_ * INNER + h * DH_, INNER,
                hout + (size_t)b * NN_ * INNER + h * DH_, INNER,
                nullptr, nullptr, 0, NN_);

    // 3b) t[b,h,i,c] = sum_j attn * gelu(e@Wev1+bev1)
    edge_value_acc<<<BB * NN_, 256, 0, stream>>>(ea, attn, Wev1, bev1, tb);

    // 3c) hout += t @ Wev2[:, h*64:(h+1)*64] + bev2  per (b,h)  (WMMA)
    for (int b = 0; b < BB; ++b)
        for (int h = 0; h < H_; ++h) {
            float* c = hout + (size_t)b * NN_ * INNER + h * DH_;
            wmma_gemm_f32<<<dim3(NN_ / 64, DH_ / 64), 128, 0, stream>>>(
                tb + (((size_t)b * H_ + h) * NN_) * INNER, INNER,
                Wev2 + h * DH_, INNER,
                c, INNER,
                bev2 + h * DH_, c, INNER, INNER);
        }

    // 4) final: out = hout @ Wo + bo   [1024,256]@[256,512]  (WMMA)
    wmma_gemm_f32<<<dim3(M / 64, QD_ / 64), 128, 0, stream>>>(
        hout, INNER, Wo, QD_, out, QD_, bo, nullptr, 0, INNER);
}